// TransformerBlockQuantum_65481071408051
// MI455X (gfx1250) — compile-verified
//
#include <hip/hip_runtime.h>
#include <math.h>

typedef __attribute__((ext_vector_type(2))) float v2f;
typedef __attribute__((ext_vector_type(8))) float v8f;

#define NQ   8
#define QDIM 256   // 2^NQ amplitudes
#define DD   8     // model dim
#define FF   512   // ffn hidden dim

// ---------------------------------------------------------------------------
// P1: exact 8-qubit statevector simulation (one block, 256 threads).
//     ws[0..7]   <- attn_out = Wc @ attn_vec + bc
//     ws[16..23] <- ffn_vec  (raw <Z> of RY circuit, consumed by P2)
// Wire w of the (2,)*8 tensor corresponds to bit (7-w) of the flat index.
// ---------------------------------------------------------------------------
__global__ void qc_prep_kernel(const float* __restrict__ rx,
                               const float* __restrict__ ry,
                               const float* __restrict__ Wc,
                               const float* __restrict__ bc,
                               float* __restrict__ ws) {
    __shared__ float sre[QDIM];
    __shared__ float simg[QDIM];
    __shared__ float zv[NQ];
    const int tid = threadIdx.x;

    // ---------------- attention circuit: RX on each wire, CNOT ring --------
    sre[tid]  = (tid == 0) ? 1.f : 0.f;
    simg[tid] = 0.f;
    __syncthreads();

    for (int w = 0; w < NQ; ++w) {
        const int   mask = 1 << (7 - w);
        const float th   = 0.5f * rx[w];
        const float c = cosf(th), s = sinf(th);
        const int   p = tid ^ mask;
        // RX: new = c*own + (-i*s)*partner
        const float r0 = sre[tid], i0 = simg[tid];
        const float r1 = sre[p],   i1 = simg[p];
        __syncthreads();
        sre[tid]  = c * r0 + s * i1;
        simg[tid] = c * i0 - s * r1;
        __syncthreads();
    }
    for (int w = 0; w < NQ; ++w) {              // CNOT(w, w+1), w=0..6; then CNOT(7,0)
        const int ctrl = w, tgt = (w + 1) & 7;
        const int cm = 1 << (7 - ctrl), tm = 1 << (7 - tgt);
        const int src = (tid & cm) ? (tid ^ tm) : tid;
        const float r = sre[src], i = simg[src];
        __syncthreads();
        sre[tid] = r; simg[tid] = i;
        __syncthreads();
    }
    {   // probabilities into sre
        const float r = sre[tid], i = simg[tid];
        __syncthreads();
        sre[tid] = r * r + i * i;
        __syncthreads();
    }
    if (tid < NQ) {                              // <Z_w> signed reductions
        const int shift = 7 - tid;
        float acc = 0.f;
        for (int idx = 0; idx < QDIM; ++idx)
            acc += ((idx >> shift) & 1) ? -sre[idx] : sre[idx];
        zv[tid] = acc;
    }
    __syncthreads();
    if (tid < DD) {                              // attn_out = Wc @ attn_vec + bc
        float acc = bc[tid];
        for (int k = 0; k < NQ; ++k) acc += Wc[tid * NQ + k] * zv[k];
        ws[tid] = acc;
    }
    __syncthreads();

    // ---------------- ffn circuit: RY on each wire (no entanglers) ---------
    sre[tid]  = (tid == 0) ? 1.f : 0.f;
    simg[tid] = 0.f;
    __syncthreads();
    for (int w = 0; w < NQ; ++w) {
        const int   mask = 1 << (7 - w);
        const float th   = 0.5f * ry[w];
        const float c = cosf(th), s = sinf(th);
        const int   p = tid ^ mask;
        const float r0 = sre[tid], i0 = simg[tid];
        const float r1 = sre[p],   i1 = simg[p];
        __syncthreads();
        const float sgn = (tid & mask) ? s : -s;  // bit=1: +s*partner, bit=0: -s*partner
        sre[tid]  = c * r0 + sgn * r1;
        simg[tid] = c * i0 + sgn * i1;
        __syncthreads();
    }
    {
        const float r = sre[tid], i = simg[tid];
        __syncthreads();
        sre[tid] = r * r + i * i;
        __syncthreads();
    }
    if (tid < NQ) {
        const int shift = 7 - tid;
        float acc = 0.f;
        for (int idx = 0; idx < QDIM; ++idx)
            acc += ((idx >> shift) & 1) ? -sre[idx] : sre[idx];
        ws[16 + tid] = acc;                      // ffn_vec
    }
}

// ---------------------------------------------------------------------------
// P2: h = relu(W1 @ ffn_vec + b1) via V_WMMA_F32_16X16X4_F32, then
//     ws[8..15] <- ffn_out = W2 @ h + b2.
// One block of 512 threads = 16 waves; wave i handles rows [32i, 32i+32) of W1
// as two 16-row tiles, each tile = two K=4 WMMAs (K=8 total). B-matrix columns
// all replicate ffn_vec, so every column of D equals the matvec result; we
// read column N=0 (lanes 0 and 16, VGPRs 0..7).
// A layout (16x4 f32): lane m=lane&15 is row M; VGPR0/1 hold K={2h,2h+1},
// h = lane>>4. B layout mirrors this in K.
// ---------------------------------------------------------------------------
__global__ void ffn_prep_kernel(const float* __restrict__ W1p,
                                const float* __restrict__ b1p,
                                const float* __restrict__ W2p,
                                const float* __restrict__ b2p,
                                float* __restrict__ ws) {
    __shared__ float h[FF];
    const int tid  = threadIdx.x;      // 0..511
    const int lane = tid & 31;
    const int wave = tid >> 5;         // 0..15
    const int m    = lane & 15;
    const int k0   = (lane >> 4) << 1; // 0 or 2
    const float* fv = ws + 16;         // ffn_vec (uniform loads, L2-cached)

    v2f bm0, bm1;
    bm0.x = fv[k0];     bm0.y = fv[k0 + 1];
    bm1.x = fv[4 + k0]; bm1.y = fv[5 + k0];

    #pragma unroll
    for (int t = 0; t < 2; ++t) {
        const int base = wave * 32 + t * 16;
        const int row  = base + m;
        v2f a0, a1;
        a0.x = W1p[row * NQ + k0];     a0.y = W1p[row * NQ + k0 + 1];
        a1.x = W1p[row * NQ + 4 + k0]; a1.y = W1p[row * NQ + 5 + k0];
        v8f acc = {};
        acc = __builtin_amdgcn_wmma_f32_16x16x4_f32(false, a0, false, bm0,
                                                    (short)0, acc, false, false);
        acc = __builtin_amdgcn_wmma_f32_16x16x4_f32(false, a1, false, bm1,
                                                    (short)0, acc, false, false);
        // Column N=0: lane 0 holds M=0..7 (VGPR r -> M=r), lane 16 holds M=8..15.
        if (lane == 0) {
            #pragma unroll
            for (int r = 0; r < 8; ++r) {
                const float v = acc[r] + b1p[base + r];
                h[base + r] = v > 0.f ? v : 0.f;
            }
        } else if (lane == 16) {
            #pragma unroll
            for (int r = 0; r < 8; ++r) {
                const float v = acc[r] + b1p[base + 8 + r];
                h[base + 8 + r] = v > 0.f ? v : 0.f;
            }
        }
    }
    __syncthreads();
    if (tid < DD) {                    // ffn_out = W2 @ h + b2 (8 x 512 matvec)
        float acc = b2p[tid];
        for (int f = 0; f < FF; ++f) acc += W2p[tid * FF + f] * h[f];
        ws[8 + tid] = acc;
    }
}

// ---------------------------------------------------------------------------
// Main streaming pass: one thread per (b,s) position (32 bytes in / 32 out).
// t  = x + attn_out ; x1 = LN1(t) ; s = x1 + ffn_out ; out = LN2(s)
// Bandwidth-bound: ~33.5 MB total @ 23.3 TB/s.
// ---------------------------------------------------------------------------
__global__ void fused_main_kernel(const float* __restrict__ x,
                                  const float* __restrict__ ln1w,
                                  const float* __restrict__ ln1b,
                                  const float* __restrict__ ln2w,
                                  const float* __restrict__ ln2b,
                                  const float* __restrict__ ws,
                                  float* __restrict__ out, int npos) {
    __shared__ float cA[8], cF[8], c1w[8], c1b[8], c2w[8], c2b[8];
    const int tid = threadIdx.x;
    if (tid < 8) {
        cA[tid]  = ws[tid];      cF[tid]  = ws[8 + tid];
        c1w[tid] = ln1w[tid];    c1b[tid] = ln1b[tid];
        c2w[tid] = ln2w[tid];    c2b[tid] = ln2b[tid];
    }
    __syncthreads();

    const int pos = blockIdx.x * blockDim.x + tid;
    if (pos >= npos) return;

    const float4* xv = reinterpret_cast<const float4*>(x) + (size_t)pos * 2;
    const float4 xa = xv[0], xb = xv[1];
    float t[8] = {xa.x, xa.y, xa.z, xa.w, xb.x, xb.y, xb.z, xb.w};

    float sum = 0.f;
    #pragma unroll
    for (int d = 0; d < 8; ++d) { t[d] += cA[d]; sum += t[d]; }
    float mean = sum * 0.125f, var = 0.f;
    #pragma unroll
    for (int d = 0; d < 8; ++d) { const float u = t[d] - mean; t[d] = u; var += u * u; }
    float r = rsqrtf(var * 0.125f + 1e-5f);

    sum = 0.f;
    #pragma unroll
    for (int d = 0; d < 8; ++d) { t[d] = t[d] * r * c1w[d] + c1b[d] + cF[d]; sum += t[d]; }
    mean = sum * 0.125f; var = 0.f;
    #pragma unroll
    for (int d = 0; d < 8; ++d) { const float u = t[d] - mean; t[d] = u; var += u * u; }
    r = rsqrtf(var * 0.125f + 1e-5f);

    float4 oa, ob;
    oa.x = t[0] * r * c2w[0] + c2b[0];
    oa.y = t[1] * r * c2w[1] + c2b[1];
    oa.z = t[2] * r * c2w[2] + c2b[2];
    oa.w = t[3] * r * c2w[3] + c2b[3];
    ob.x = t[4] * r * c2w[4] + c2b[4];
    ob.y = t[5] * r * c2w[5] + c2b[5];
    ob.z = t[6] * r * c2w[6] + c2b[6];
    ob.w = t[7] * r * c2w[7] + c2b[7];
    float4* ov = reinterpret_cast<float4*>(out) + (size_t)pos * 2;
    ov[0] = oa;
    ov[1] = ob;
}

extern "C" void kernel_launch(void* const* d_in, const int* in_sizes, int n_in,
                              void* d_out, int out_size, void* d_ws, size_t ws_size,
                              hipStream_t stream) {
    const float* x   = (const float*)d_in[0];
    const float* rx  = (const float*)d_in[1];
    const float* ry  = (const float*)d_in[2];
    const float* Wc  = (const float*)d_in[3];
    const float* bc  = (const float*)d_in[4];
    const float* W1  = (const float*)d_in[5];
    const float* b1  = (const float*)d_in[6];
    const float* W2  = (const float*)d_in[7];
    const float* b2  = (const float*)d_in[8];
    const float* l1w = (const float*)d_in[9];
    const float* l1b = (const float*)d_in[10];
    const float* l2w = (const float*)d_in[11];
    const float* l2b = (const float*)d_in[12];
    float* out = (float*)d_out;
    float* ws  = (float*)d_ws;   // [0..7]=attn_out [8..15]=ffn_out [16..23]=ffn_vec

    const int npos = in_sizes[0] / DD;  // 64*8192 positions

    qc_prep_kernel<<<1, QDIM, 0, stream>>>(rx, ry, Wc, bc, ws);
    ffn_prep_kernel<<<1, FF, 0, stream>>>(W1, b1, W2, b2, ws);
    const int blocks = (npos + 255) / 256;
    fused_main_kernel<<<blocks, 256, 0, stream>>>(x, l1w, l1b, l2w, l2b, ws, out, npos);
}